// PairwiseMaxMarginHingeLossFunc_21775484190866
// MI455X (gfx1250) — compile-verified
//
#include <hip/hip_runtime.h>

// CDNA5 / gfx1250, wave32. One wave computes the full 64x64 hinge tile for one
// (batch, row) pair using V_WMMA_F32_16X16X4_F32: the pairwise product matrix
// (o_j - o_k)(l_j - l_k) is rank-4, so  margin - d*l = A' x B + margin_C  with
//   A'_j = [-o_j*l_j, -1, o_j, l_j]   (16x4 tile per WMMA)
//   B_k  = [ 1, o_k*l_k, l_k, o_k ]   (4x16 tile per WMMA)
// 16 WMMAs cover the 64x64 matrix; VALU only does relu+accumulate.

typedef __attribute__((ext_vector_type(2))) float v2f;
typedef __attribute__((ext_vector_type(8))) float v8f;

#define N_DIM 64
#define BPW   8   // batches per workgroup == waves per 256-thread block

__global__ void hinge_init_kernel(float* out, int out_size) {
    int t = threadIdx.x;
    if (t < out_size) out[t] = 0.0f;
}

__global__ __launch_bounds__(256) void hinge_wmma_kernel(
    const float* __restrict__ out_m,   // [n, n, B]
    const float* __restrict__ lab_m,   // [n, n, B]
    const float* __restrict__ margin_p,
    float* __restrict__ out,           // out[0]=loss, out[1]=correct count
    int B)
{
    // Stage one row (64 elems) x 8 batches of both matrices through LDS.
    // Pad inner dim to 9 floats to avoid bank conflicts on the strided reads.
    __shared__ float so[N_DIM][BPW + 1];
    __shared__ float sl[N_DIM][BPW + 1];

    const int tid  = threadIdx.x;
    const int lane = tid & 31;
    const int wave = tid >> 5;              // 0..7 -> batch within group
    const int nbg  = B / BPW;
    const int row  = blockIdx.x / nbg;      // i in [0, n)
    const int b0   = (blockIdx.x % nbg) * BPW;

    // Coalesced-ish load: 8 consecutive threads fetch 8 consecutive batches
    // (32B runs) of element (row, j, b0+bb).
    #pragma unroll
    for (int e = tid; e < N_DIM * BPW; e += 256) {
        const int j  = e >> 3;
        const int bb = e & 7;
        const size_t g = ((size_t)(row * N_DIM + j)) * (size_t)B + (size_t)(b0 + bb);
        so[j][bb] = out_m[g];
        sl[j][bb] = lab_m[g];
    }
    __syncthreads();

    const float margin = *margin_p;
    const int   hl     = lane & 15;
    const bool  hi     = lane >= 16;

    // Each lane needs (o, l) at j = g*16 + (lane&15) for the 4 tile groups.
    float o4[4], l4[4];
    #pragma unroll
    for (int g = 0; g < 4; ++g) {
        o4[g] = so[g * 16 + hl][wave];
        l4[g] = sl[g * 16 + hl][wave];
    }

    // Build A'/B fragments per 16-wide group.
    // A 16x4 f32 layout: lanes 0-15 hold K=0 (v0) / K=1 (v1) for M=lane;
    //                    lanes 16-31 hold K=2 / K=3 for M=lane-16.
    // B 4x16 f32 layout mirrors it (rows striped across lanes per VGPR).
    v2f afr[4], bfr[4];
    #pragma unroll
    for (int g = 0; g < 4; ++g) {
        const float o = o4[g], l = l4[g];
        afr[g].x = hi ? o : (-o * l);   // K=2: o_j      | K=0: -o_j*l_j
        afr[g].y = hi ? l : -1.0f;      // K=3: l_j      | K=1: -1
        bfr[g].x = hi ? l : 1.0f;       // K=2: l_k      | K=0:  1
        bfr[g].y = hi ? o : (o * l);    // K=3: o_k      | K=1:  o_k*l_k
    }

    v8f cmargin;
    #pragma unroll
    for (int e = 0; e < 8; ++e) cmargin[e] = margin;

    // 16 WMMAs tile the 64x64 hinge-argument matrix: D = margin - d*l.
    float loss_acc = 0.0f;
    #pragma unroll
    for (int tm = 0; tm < 4; ++tm) {
        #pragma unroll
        for (int tn = 0; tn < 4; ++tn) {
            v8f d = __builtin_amdgcn_wmma_f32_16x16x4_f32(
                /*neg_a=*/false, afr[tm],
                /*neg_b=*/false, bfr[tn],
                /*c_mod=*/(short)0, cmargin,
                /*reuse_a=*/false, /*reuse_b=*/false);
            #pragma unroll
            for (int e = 0; e < 8; ++e)
                loss_acc += fmaxf(d[e], 0.0f);
        }
    }

    // ---- argmax agreement (first-index tie break) ----
    float bo = o4[0], bl = l4[0];
    int   jo = hl,    jl = hl;
    #pragma unroll
    for (int g = 1; g < 4; ++g) {
        const int j = g * 16 + hl;
        if (o4[g] > bo) { bo = o4[g]; jo = j; }
        if (l4[g] > bl) { bl = l4[g]; jl = j; }
    }
    // wave32 xor-reduction (lanes 16-31 hold duplicates of 0-15: harmless)
    #pragma unroll
    for (int m = 16; m >= 1; m >>= 1) {
        const float oo = __shfl_xor(bo, m, 32);
        const int  ojo = __shfl_xor(jo, m, 32);
        if (oo > bo || (oo == bo && ojo < jo)) { bo = oo; jo = ojo; }
        const float ol = __shfl_xor(bl, m, 32);
        const int  ojl = __shfl_xor(jl, m, 32);
        if (ol > bl || (ol == bl && ojl < jl)) { bl = ol; jl = ojl; }
    }

    // ---- loss reduction across the wave ----
    #pragma unroll
    for (int m = 16; m >= 1; m >>= 1)
        loss_acc += __shfl_xor(loss_acc, m, 32);

    if (lane == 0) {
        atomicAdd(&out[0], loss_acc);
        if (jo == jl) atomicAdd(&out[1], 1.0f);
    }
}

extern "C" void kernel_launch(void* const* d_in, const int* in_sizes, int n_in,
                              void* d_out, int out_size, void* d_ws, size_t ws_size,
                              hipStream_t stream) {
    const float* out_m  = (const float*)d_in[0];
    const float* lab_m  = (const float*)d_in[1];
    const float* margin = (const float*)d_in[2];
    float* out = (float*)d_out;

    const int B   = in_sizes[0] / (N_DIM * N_DIM);   // 256 for the reference
    const int nbg = B / BPW;

    hinge_init_kernel<<<1, 64, 0, stream>>>(out, out_size);
    hinge_wmma_kernel<<<dim3(N_DIM * nbg), 256, 0, stream>>>(out_m, lab_m, margin, out, B);
}